// SVDAlign_30932354466352
// MI455X (gfx1250) — compile-verified
//
#include <hip/hip_runtime.h>
#include <math.h>

// ---------------------------------------------------------------------------
// Weighted Kabsch alignment, fused single-pass kernel for MI455X (gfx1250).
// One block (256 thr = 8 waves) handles 4 batches. Memory-bound: ~105 MB
// traffic -> ~4.5 us floor at 23.3 TB/s; all phases fused so input is
// streamed once (apply-phase re-read hits the 192 MB L2).
//  Phase 1: per-point derived values -> LDS  (wa2 = w*(v2,1), a1 = (v1,1)),
//           points padded 200->224 with zeros.
//  Phase 2: S4[b] = sum_n wa2_n a1_n^T via V_WMMA_F32_16X16X4_F32,
//           4 batches packed as diagonal 4x4 blocks of the 16x16 tile,
//           56 K-chunks = exactly 7 per wave (fully unrolled, pipelined),
//           tree-combined through LDS (deterministic).
//  Phase 3: one lane per batch: cov from S4, 3x3 Jacobi SVD, Kabsch R,t.
//  Phase 4: new = R*xyz2 + t streamed to global.
// ---------------------------------------------------------------------------

typedef __attribute__((ext_vector_type(2))) float v2f;
typedef __attribute__((ext_vector_type(8))) float v8f;

#ifndef __has_builtin
#define __has_builtin(x) 0
#endif
#if __has_builtin(__builtin_amdgcn_wmma_f32_16x16x4_f32)
#define USE_WMMA 1
#else
#define USE_WMMA 0
#endif

#define NPTS 200
#define NPAD 224                 // padded point count: 56 chunks = 7 per wave
#define BPB  4                   // batches per block (diag 4x4 blocks of 16x16)
#define NTHR 256
#define NWAVE (NTHR/32)
#define CHUNKS_PER_WAVE (NPAD/4/NWAVE)   // 7

// Jacobi rotation on symmetric Mm, accumulating eigenvectors in V (M = V D V^T)
#define JROT(P,Q,RR)                                                          \
  { float apq = Mm[P][Q];                                                     \
    if (fabsf(apq) > 1e-30f) {                                                \
      float th = (Mm[Q][Q] - Mm[P][P]) / (2.0f*apq);                          \
      float tt = copysignf(1.0f, th) / (fabsf(th) + sqrtf(fmaf(th,th,1.0f))); \
      float cc = 1.0f / sqrtf(fmaf(tt,tt,1.0f));                              \
      float ss = tt*cc;                                                       \
      Mm[P][P] -= tt*apq;                                                     \
      Mm[Q][Q] += tt*apq;                                                     \
      Mm[P][Q] = 0.0f; Mm[Q][P] = 0.0f;                                       \
      float mrp = Mm[RR][P], mrq = Mm[RR][Q];                                 \
      Mm[RR][P] = cc*mrp - ss*mrq; Mm[P][RR] = Mm[RR][P];                     \
      Mm[RR][Q] = ss*mrp + cc*mrq; Mm[Q][RR] = Mm[RR][Q];                     \
      float a0p=V[0][P], a0q=V[0][Q]; V[0][P]=cc*a0p-ss*a0q; V[0][Q]=ss*a0p+cc*a0q; \
      float a1p=V[1][P], a1q=V[1][Q]; V[1][P]=cc*a1p-ss*a1q; V[1][Q]=ss*a1p+cc*a1q; \
      float a2p=V[2][P], a2q=V[2][Q]; V[2][P]=cc*a2p-ss*a2q; V[2][Q]=ss*a2p+cc*a2q; \
    } }

__global__ __launch_bounds__(NTHR) void svd_align_kernel(
    const float* __restrict__ net,     // [B][200][13]
    const float* __restrict__ pshift,  // [1]
    const float* __restrict__ pa,      // [1]
    const float* __restrict__ pb,      // [1]
    float* __restrict__ out,           // [B][200][3]
    int B)
{
  // [point][m] with m = bi*4 + comp  (16 floats per point -> conflict-free
  // WMMA fragment gathers: lanes 0..15 hit banks 0..15 / 32..47)
  __shared__ alignas(16) float wa2f[NPAD*16];   // w * (v2x,v2y,v2z,1)
  __shared__ alignas(16) float a1f [NPAD*16];   // (v1x,v1y,v1z,1)
#if USE_WMMA
  __shared__ float red[NWAVE][32][9];           // per-wave D tiles (padded: 9)
#endif
  __shared__ float S4[BPB][4][4];               // per-batch 4x4 moment matrix
  __shared__ float Rm[BPB][3][3];
  __shared__ float Tv[BPB][3];

  const int tid  = threadIdx.x;
  const int b0   = blockIdx.x * BPB;
  const float shift = *pshift;
  const float aa    = *pa;
  const float bb    = *pb;

  // -------- Phase 1: per-point derived values into LDS (zero padding) ------
  for (int idx = tid; idx < BPB*NPAD; idx += NTHR) {
    const int bi = idx / NPAD;
    const int p  = idx - bi*NPAD;
    const int b  = b0 + bi;
    float w = 0.f, v1x=0.f, v1y=0.f, v1z=0.f, v2x=0.f, v2y=0.f, v2z=0.f, one=0.f;
    if (b < B && p < NPTS) {
      const float* q = net + ((size_t)b*NPTS + p)*13;
      float w0 = q[0];
      w   = fmaxf(fmaf(aa, w0, bb), 0.f) + 1e-8f;
      v1x = fmaf(q[7],  shift, q[1]);
      v1y = fmaf(q[8],  shift, q[2]);
      v1z = fmaf(q[9],  shift, q[3]);
      v2x = fmaf(q[10], shift, q[4]);
      v2y = fmaf(q[11], shift, q[5]);
      v2z = fmaf(q[12], shift, q[6]);
      one = 1.f;
    }
    float4 wv = make_float4(w*v2x, w*v2y, w*v2z, w);
    float4 av = make_float4(v1x, v1y, v1z, one);
    *reinterpret_cast<float4*>(&wa2f[p*16 + bi*4]) = wv;
    *reinterpret_cast<float4*>(&a1f [p*16 + bi*4]) = av;
  }
  __syncthreads();

  // -------- Phase 2: moment accumulation --------
#if USE_WMMA
  {
    // scalar (SGPR) wave id -> uniform loop control, EXEC untouched
    const int wave = __builtin_amdgcn_readfirstlane(tid >> 5);
    const int lane = tid & 31;
    const int m16  = lane & 15;             // row index (m) == col index (n)
    const int koff = (lane >= 16) ? 2 : 0;  // A/B VGPR0 holds K=0 (lo) / K=2 (hi)
    const int base = 16*koff + m16;         // per-lane fragment base (floats)

    v8f acc = {0.f,0.f,0.f,0.f,0.f,0.f,0.f,0.f};
    // 56 K-chunks of 4 points, exactly 7 per wave; fully unrolled so all
    // LDS fragment loads issue ahead of the WMMA chain (latency hidden).
#pragma unroll
    for (int k = 0; k < CHUNKS_PER_WAVE; ++k) {
      const int c  = wave + NWAVE*k;        // uniform scalar chunk id
      const int a0 = 64*c + base;           // element offset of K=koff point
      v2f Af, Bf;
      Af.x = wa2f[a0];
      Af.y = wa2f[a0 + 16];
      Bf.x = a1f [a0];
      Bf.y = a1f [a0 + 16];
      acc = __builtin_amdgcn_wmma_f32_16x16x4_f32(
          /*neg_a=*/false, Af, /*neg_b=*/false, Bf,
          /*c_mod=*/(short)0, acc, /*reuse_a=*/false, /*reuse_b=*/false);
    }
#pragma unroll
    for (int k = 0; k < 8; ++k) red[wave][lane][k] = acc[k];
  }
  __syncthreads();

  // Combine the 8 per-wave partial D tiles; extract diagonal 4x4 blocks.
  if (tid < BPB*16) {
    const int bi = tid >> 4;
    const int e  = tid & 15;
    const int i  = e >> 2;          // row within block
    const int j  = e & 3;           // col within block
    const int m  = bi*4 + i;        // global tile row
    const int n  = bi*4 + j;        // global tile col
    const int lane = (m >= 8) ? (n + 16) : n;   // D layout: vgpr v <-> M=v / v+8
    const int v    = m & 7;
    float s = 0.f;
#pragma unroll
    for (int wv = 0; wv < NWAVE; ++wv) s += red[wv][lane][v];
    S4[bi][i][j] = s;
  }
#else
  // VALU fallback: direct dot products (only used if builtin is absent)
  if (tid < BPB*16) {
    const int bi = tid >> 4;
    const int e  = tid & 15;
    const int i  = e >> 2;
    const int j  = e & 3;
    float s = 0.f;
    for (int p = 0; p < NPTS; ++p)
      s += wa2f[p*16 + bi*4 + i] * a1f[p*16 + bi*4 + j];
    S4[bi][i][j] = s;
  }
#endif
  __syncthreads();

  // -------- Phase 3: per-batch 3x3 Kabsch SVD (one lane each) --------
  if (tid < BPB) {
    const int bi = tid;
    const float wsum = S4[bi][3][3];
    const float inv  = 1.f / wsum;
    float swv1[3] = { S4[bi][3][0], S4[bi][3][1], S4[bi][3][2] };
    float swv2[3] = { S4[bi][0][3], S4[bi][1][3], S4[bi][2][3] };
    float v1c[3], v2c[3];
#pragma unroll
    for (int k = 0; k < 3; ++k) { v1c[k] = swv1[k]*inv; v2c[k] = swv2[k]*inv; }

    // centered covariance: cov = S - swv2 * swv1^T / wsum
    float C[3][3];
#pragma unroll
    for (int i = 0; i < 3; ++i)
#pragma unroll
      for (int j = 0; j < 3; ++j)
        C[i][j] = S4[bi][i][j] - swv2[i]*swv1[j]*inv;

    // Mm = C^T C (symmetric)
    float Mm[3][3];
#pragma unroll
    for (int i = 0; i < 3; ++i)
#pragma unroll
      for (int j = 0; j < 3; ++j)
        Mm[i][j] = C[0][i]*C[0][j] + C[1][i]*C[1][j] + C[2][i]*C[2][j];

    float V[3][3] = {{1.f,0.f,0.f},{0.f,1.f,0.f},{0.f,0.f,1.f}};
#pragma unroll
    for (int sweep = 0; sweep < 6; ++sweep) {
      JROT(0,1,2); JROT(0,2,1); JROT(1,2,0);
    }

    const float eig[3] = { Mm[0][0], Mm[1][1], Mm[2][2] };
    const float det =
        C[0][0]*(C[1][1]*C[2][2] - C[1][2]*C[2][1])
      - C[0][1]*(C[1][0]*C[2][2] - C[1][2]*C[2][0])
      + C[0][2]*(C[1][0]*C[2][1] - C[1][1]*C[2][0]);
    const float dsgn = (det < 0.f) ? -1.f : 1.f;   // det(V U^T)

    int kmin = 0; float emin = eig[0];
    if (eig[1] < emin) { kmin = 1; emin = eig[1]; }
    if (eig[2] < emin) { kmin = 2; }

    float coef[3];
#pragma unroll
    for (int k = 0; k < 3; ++k) {
      float sv = sqrtf(fmaxf(eig[k], 0.f));
      float ic = 1.f / fmaxf(sv, 1e-20f);
      coef[k] = (k == kmin) ? dsgn*ic : ic;
    }

    // CV[:,k] = C * v_k  (= s_k * u_k);  R = sum_k coef_k v_k (C v_k)^T
    float CV[3][3];
#pragma unroll
    for (int j = 0; j < 3; ++j)
#pragma unroll
      for (int k = 0; k < 3; ++k)
        CV[j][k] = C[j][0]*V[0][k] + C[j][1]*V[1][k] + C[j][2]*V[2][k];

    float Rl[3][3];
#pragma unroll
    for (int i = 0; i < 3; ++i)
#pragma unroll
      for (int j = 0; j < 3; ++j) {
        Rl[i][j] = coef[0]*V[i][0]*CV[j][0]
                 + coef[1]*V[i][1]*CV[j][1]
                 + coef[2]*V[i][2]*CV[j][2];
        Rm[bi][i][j] = Rl[i][j];
      }
#pragma unroll
    for (int i = 0; i < 3; ++i)
      Tv[bi][i] = v1c[i] - (Rl[i][0]*v2c[0] + Rl[i][1]*v2c[1] + Rl[i][2]*v2c[2]);
  }
  __syncthreads();

  // -------- Phase 4: apply R,t to raw xyz2, stream out --------
  for (int idx = tid; idx < BPB*NPTS; idx += NTHR) {
    const int bi = idx / NPTS;
    const int p  = idx - bi*NPTS;
    const int b  = b0 + bi;
    if (b >= B) continue;
    const float* q = net + ((size_t)b*NPTS + p)*13;   // L2-resident re-read
    const float x = q[4], y = q[5], z = q[6];
    float* o = out + ((size_t)b*NPTS + p)*3;
    o[0] = Rm[bi][0][0]*x + Rm[bi][0][1]*y + Rm[bi][0][2]*z + Tv[bi][0];
    o[1] = Rm[bi][1][0]*x + Rm[bi][1][1]*y + Rm[bi][1][2]*z + Tv[bi][1];
    o[2] = Rm[bi][2][0]*x + Rm[bi][2][1]*y + Rm[bi][2][2]*z + Tv[bi][2];
  }
}

extern "C" void kernel_launch(void* const* d_in, const int* in_sizes, int n_in,
                              void* d_out, int out_size, void* d_ws, size_t ws_size,
                              hipStream_t stream) {
  (void)n_in; (void)out_size; (void)d_ws; (void)ws_size;
  const float* net    = (const float*)d_in[0];
  const float* pshift = (const float*)d_in[1];
  const float* pa     = (const float*)d_in[2];
  const float* pb     = (const float*)d_in[3];
  float* out = (float*)d_out;
  const int B = in_sizes[0] / (NPTS * 13);
  const int nblocks = (B + BPB - 1) / BPB;
  svd_align_kernel<<<dim3(nblocks), dim3(NTHR), 0, stream>>>(
      net, pshift, pa, pb, out, B);
}